// CRF_11879879544115
// MI455X (gfx1250) — compile-verified
//
#include <hip/hip_runtime.h>
#include <stdint.h>

typedef __attribute__((ext_vector_type(2))) float v2f;
typedef __attribute__((ext_vector_type(8))) float v8f;

#define B_   4096
#define T_   512
#define N_   48
#define WPB  8               // waves (batches) per block
#define BLOCK_ (WPB * 32)
#define CHK  64              // backtrack staging chunk (timesteps)

// CDNA5 async global->LDS copy (ASYNCcnt-tracked), GVS addressing.
__device__ __forceinline__ void async_load_b128(uint32_t lds_off, const void* base, uint32_t goff) {
  asm volatile("global_load_async_to_lds_b128 %0, %1, %2"
               :: "v"(lds_off), "v"(goff), "s"(base)
               : "memory");
}
__device__ __forceinline__ void wait_async0() {
  asm volatile("s_wait_asynccnt 0" ::: "memory");
}

// Half-wave swap (lane <-> lane^16) as a single VALU op: v_permlanex16_b32
// with identity selectors (nibble i of sel0 = i, of sel1 = 8+i).
__device__ __forceinline__ int swap16_i(int x) {
  return __builtin_amdgcn_permlanex16(x, x, 0x76543210, 0xFEDCBA98, false, false);
}
__device__ __forceinline__ float swap16_f(float x) {
  int i = __builtin_bit_cast(int, x);
  i = __builtin_amdgcn_permlanex16(i, i, 0x76543210, 0xFEDCBA98, false, false);
  return __builtin_bit_cast(float, i);
}

__global__ __launch_bounds__(BLOCK_) void crf_viterbi_wmma(
    const float* __restrict__ inputs,     // [B,T,N]
    const float* __restrict__ trans,      // [N,N]
    const int*   __restrict__ seq_lens,   // [B]
    float* __restrict__ out,              // [B,T,N] one-hot
    uint8_t* __restrict__ bp)             // [B,T,N] backpointers (workspace)
{
  __shared__ float s_trans[N_ * N_];                        // 9216 B
  __shared__ float s_state[WPB][64];                        // final-state scratch
  __shared__ __align__(16) float   s_pot[WPB][2][N_];       // emission double buffer
  __shared__ __align__(16) uint8_t s_bp[WPB][2][CHK * N_];  // backtrack staging (2x3072 B/wave)

  const int tid  = threadIdx.x;
  const int wave = tid >> 5;
  const int lane = tid & 31;
  const int half = lane >> 4;
  const int li   = lane & 15;
  const int b    = blockIdx.x * WPB + wave;

  for (int k = tid; k < N_ * N_; k += BLOCK_) s_trans[k] = trans[k];
  __syncthreads();

  // Transition tiles in WMMA C/D layout: cfrag[nj][mi][v] = trans[mi*16+8*half+v][nj*16+li]
  v8f cfrag[3][3];
  #pragma unroll
  for (int nj = 0; nj < 3; ++nj)
    #pragma unroll
    for (int mi = 0; mi < 3; ++mi)
      #pragma unroll
      for (int v = 0; v < 8; ++v)
        cfrag[nj][mi][v] = s_trans[(mi * 16 + half * 8 + v) * N_ + nj * 16 + li];

  // B = 4x16 with ones in row K=0 (lanes 0-15 VGPR0).
  v2f bfrag; bfrag.x = (half == 0) ? 1.0f : 0.0f; bfrag.y = 0.0f;

  const int     slen  = max(seq_lens[b], 1);
  const float*  inp_b = inputs + (size_t)b * T_ * N_;
  const size_t  base  = (size_t)b * T_ * N_;

  // state layout: st0 = state[lane], st1 = state[32+li]
  float st0 = inp_b[lane];
  float st1 = inp_b[32 + li];

  const uint32_t pot_lds0 = (uint32_t)(uintptr_t)(&s_pot[wave][0][0]);
  const uint32_t pot_lds1 = (uint32_t)(uintptr_t)(&s_pot[wave][1][0]);
  if (lane < 12) {
    async_load_b128(pot_lds0 + lane * 16, inp_b, (uint32_t)(1 * N_ * 4 + lane * 16));
  }

  int buf = 0;
  for (int t = 1; t < T_; ++t) {
    wait_async0();                                   // emissions for step t are in LDS
    if ((t + 1 < T_) && lane < 12) {
      uint32_t dst = (buf ? pot_lds0 : pot_lds1) + lane * 16;
      async_load_b128(dst, inp_b, (uint32_t)((t + 1) * N_ * 4 + lane * 16));
    }
    const float* pot = buf ? &s_pot[wave][1][0] : &s_pot[wave][0][0];

    // A fragments (16x4 f32, state in K=0 column).
    float xst = swap16_f(st0);                       // state[lane^16]
    v2f a0, a1, a2;
    a0.x = half ? 0.0f : st0;  a0.y = 0.0f;          // rows 0..15
    a1.x = half ? 0.0f : xst;  a1.y = 0.0f;          // rows 16..31
    a2.x = half ? 0.0f : st1;  a2.y = 0.0f;          // rows 32..47

    float nbest[3]; int narg[3];
    #pragma unroll
    for (int nj = 0; nj < 3; ++nj) {
      // D[m,n] = state[m]*1 + trans[m,n]  (bit-exact fma via WMMA)
      v8f d0 = __builtin_amdgcn_wmma_f32_16x16x4_f32(false, a0, false, bfrag, (short)0, cfrag[nj][0], false, false);
      v8f d1 = __builtin_amdgcn_wmma_f32_16x16x4_f32(false, a1, false, bfrag, (short)0, cfrag[nj][1], false, false);
      v8f d2 = __builtin_amdgcn_wmma_f32_16x16x4_f32(false, a2, false, bfrag, (short)0, cfrag[nj][2], false, false);

      // first-max scan over this lane's 24 rows (ascending, strict >)
      float bb = d0[0]; int br = half * 8;
      #pragma unroll
      for (int v = 1; v < 8; ++v) { float x = d0[v]; int r = half * 8 + v;      if (x > bb) { bb = x; br = r; } }
      #pragma unroll
      for (int v = 0; v < 8; ++v) { float x = d1[v]; int r = 16 + half * 8 + v; if (x > bb) { bb = x; br = r; } }
      #pragma unroll
      for (int v = 0; v < 8; ++v) { float x = d2[v]; int r = 32 + half * 8 + v; if (x > bb) { bb = x; br = r; } }

      // combine lane pair (l, l^16): strictly larger wins, tie -> smaller row
      float ob = swap16_f(bb);
      int   orow = swap16_i(br);
      bool take = (ob > bb) || ((ob == bb) && (orow < br));
      nbest[nj] = take ? ob : bb;
      narg[nj]  = take ? orow : br;
    }

    const bool valid = (t < slen);
    float old0 = half ? xst : st0;       // state[li]
    float old1 = half ? st0 : xst;       // state[16+li]
    float old2 = st1;                    // state[32+li]
    float r0 = valid ? (nbest[0] + pot[li])      : old0;
    float r1 = valid ? (nbest[1] + pot[16 + li]) : old1;
    float r2 = valid ? (nbest[2] + pot[32 + li]) : old2;
    int g0 = valid ? narg[0] : li;
    int g1 = valid ? narg[1] : (16 + li);
    int g2 = valid ? narg[2] : (32 + li);

    const size_t row = base + (size_t)t * N_;
    if (half == 0) {
      bp[row + li]      = (uint8_t)g0;
      bp[row + 16 + li] = (uint8_t)g1;
      bp[row + 32 + li] = (uint8_t)g2;
    }

    st0 = half ? r1 : r0;
    st1 = r2;
    buf ^= 1;
  }

  // final state -> LDS, first-max argmax (broadcast reads; all lanes agree)
  s_state[wave][lane] = st0;
  if (half == 0) s_state[wave][32 + li] = st1;
  __syncthreads();

  float bmax = s_state[wave][0];
  int   tag  = 0;
  #pragma unroll
  for (int i = 1; i < N_; ++i) {
    float v = s_state[wave][i];
    if (v > bmax) { bmax = v; tag = i; }
  }

  // bp stores must be L2-visible before async staging reads them back
  asm volatile("s_wait_storecnt 0" ::: "memory");

  const int  j0  = lane;
  const int  j1  = 32 + li;
  const bool hj1 = (half == 0);
  const uint8_t* bp_b = bp + base;
  const uint32_t bp_lds0 = (uint32_t)(uintptr_t)(&s_bp[wave][0][0]);
  const uint32_t bp_lds1 = (uint32_t)(uintptr_t)(&s_bp[wave][1][0]);

  // Stage last chunk (t in [T-CHK, T)): 3072 B = 6 x b128 per lane, async.
  #pragma unroll
  for (int k = 0; k < 6; ++k) {
    uint32_t off = (uint32_t)((k * 32 + lane) * 16);
    async_load_b128(bp_lds0 + off, bp_b, (uint32_t)((T_ - CHK) * N_) + off);
  }

  int bbuf = 0;
  for (int c = T_ / CHK - 1; c >= 0; --c) {
    wait_async0();                                   // chunk c staged
    if (c > 0) {                                     // prefetch chunk c-1 into other buffer
      uint32_t dst = bbuf ? bp_lds0 : bp_lds1;
      #pragma unroll
      for (int k = 0; k < 6; ++k) {
        uint32_t off = (uint32_t)((k * 32 + lane) * 16);
        async_load_b128(dst + off, bp_b, (uint32_t)((c - 1) * CHK * N_) + off);
      }
    }
    const uint8_t* cb = bbuf ? &s_bp[wave][1][0] : &s_bp[wave][0][0];
    const int tlo  = c * CHK;
    const int tend = (tlo > 0) ? tlo : 1;
    for (int t = tlo + CHK - 1; t >= tend; --t) {
      const int otag = (t < slen) ? tag : 0;
      const size_t row = base + (size_t)t * N_;
      out[row + j0] = (j0 == otag) ? 1.0f : 0.0f;
      if (hj1) out[row + j1] = (j1 == otag) ? 1.0f : 0.0f;
      tag = (int)cb[(t - tlo) * N_ + tag];           // LDS-latency dependent chain
    }
    bbuf ^= 1;
  }
  out[base + j0] = (j0 == tag) ? 1.0f : 0.0f;
  if (hj1) out[base + j1] = (j1 == tag) ? 1.0f : 0.0f;
}

extern "C" void kernel_launch(void* const* d_in, const int* in_sizes, int n_in,
                              void* d_out, int out_size, void* d_ws, size_t ws_size,
                              hipStream_t stream) {
  (void)in_sizes; (void)n_in; (void)out_size; (void)ws_size;
  const float* inputs = (const float*)d_in[0];
  const float* trans  = (const float*)d_in[1];
  const int*   lens   = (const int*)d_in[2];
  float*   out = (float*)d_out;
  uint8_t* bp  = (uint8_t*)d_ws;    // needs B*T*N = 100,663,296 bytes

  dim3 grid(B_ / WPB);              // 512 blocks x 256 threads = 4096 waves (1 batch/wave)
  crf_viterbi_wmma<<<grid, BLOCK_, 0, stream>>>(inputs, trans, lens, out, bp);
}